// Zlayer_85091892068830
// MI455X (gfx1250) — compile-verified
//
#include <hip/hip_runtime.h>
#include <hip/hip_bf16.h>

typedef __attribute__((ext_vector_type(2))) float v2f;
typedef __attribute__((ext_vector_type(8))) float v8f;

#define TT 2048
#define BB 64
#define DD 512
#define ROWS (TT * BB)          // 131072 flattened (t,b) rows
#define WAVES_PER_BLOCK 8       // 256 threads, wave32
#define ROWS_PER_BLOCK (WAVES_PER_BLOCK * 16)

// ---------------------------------------------------------------------------
// Phase 1: xproj[r] = dot(x[r, 0:512], W[0:512])  via V_WMMA_F32_16X16X4_F32
// One wave handles 16 rows. B-matrix columns all broadcast W[k..k+3], so
// every D column equals the desired dot product; lanes 0 and 16 write out
// rows 0-7 / 8-15 of the tile with b128 stores.
// ---------------------------------------------------------------------------
__global__ void __launch_bounds__(256)
xproj_wmma_kernel(const float* __restrict__ x,
                  const float* __restrict__ W,
                  float* __restrict__ xproj) {
    __shared__ float wlds[DD];
    // cooperative stage of W into LDS (removes 128 redundant global loads/lane)
    for (int i = threadIdx.x; i < DD; i += blockDim.x) wlds[i] = W[i];
    __syncthreads();

    const int lane    = threadIdx.x & 31;
    const int wave    = threadIdx.x >> 5;
    const int m       = lane & 15;        // row within tile (both lane halves)
    const int halfSel = lane >> 4;        // 0: K={k,k+1}, 1: K={k+2,k+3}

    const long tile    = (long)blockIdx.x * WAVES_PER_BLOCK + wave;
    const long rowBase = tile * 16;

    const float* xRow = x + (rowBase + m) * DD + 2 * halfSel;
    const float* wPtr = wlds + 2 * halfSel;

    v8f c = {};
#pragma unroll 8
    for (int k = 0; k < DD; k += 4) {
        v2f a = *(const v2f*)(xRow + k);      // global_load_b64, per-lane row
        v2f b = *(const v2f*)(wPtr + k);      // ds_load_b64 (broadcast per half)
        // (neg_a, A, neg_b, B, c_mod, C, reuse_a, reuse_b)
        c = __builtin_amdgcn_wmma_f32_16x16x4_f32(
                false, a, false, b, (short)0, c, false, false);
    }

    // D[m,n] identical across n. Lanes 0-15: VGPR j = row j; lanes 16-31: row 8+j.
    if (m == 0) {
        float* dst = xproj + rowBase + 8 * halfSel;
        *(float4*)(dst + 0) = make_float4(c[0], c[1], c[2], c[3]);
        *(float4*)(dst + 4) = make_float4(c[4], c[5], c[6], c[7]);
    }
}

// ---------------------------------------------------------------------------
// Phase 2: sequential sigmoid scan over t, parallel across B=64 columns.
// Register double-buffer 16 steps deep so the ~300-cycle L2 load latency
// hides behind 16 dependent fma->exp->rcp chain steps.
// ---------------------------------------------------------------------------
#define PF 16

__global__ void __launch_bounds__(64)
scan_kernel(const float* __restrict__ xproj,
            const float* __restrict__ W,
            const float* __restrict__ bptr,
            float* __restrict__ out) {
    const int   col  = threadIdx.x;       // 0..63
    const float wz   = W[DD];
    const float bias = bptr[0];

    float buf[PF], nbuf[PF];
#pragma unroll
    for (int i = 0; i < PF; ++i) buf[i] = xproj[i * BB + col];

    float z = 0.0f;
    const int ntiles = TT / PF;
    for (int tile = 0; tile < ntiles; ++tile) {
        if (tile + 1 < ntiles) {
            const int nt = (tile + 1) * PF;
#pragma unroll
            for (int i = 0; i < PF; ++i) nbuf[i] = xproj[(nt + i) * BB + col];
        }
        const int t0 = tile * PF;
#pragma unroll
        for (int i = 0; i < PF; ++i) {
            float u = buf[i] + bias + wz * z;
            z = 1.0f / (1.0f + __expf(-u));
            out[(t0 + i) * BB + col] = z;
        }
#pragma unroll
        for (int i = 0; i < PF; ++i) buf[i] = nbuf[i];
    }
}

// ---------------------------------------------------------------------------
extern "C" void kernel_launch(void* const* d_in, const int* in_sizes, int n_in,
                              void* d_out, int out_size, void* d_ws, size_t ws_size,
                              hipStream_t stream) {
    const float* x  = (const float*)d_in[0];   // (T, B, D) fp32
    const float* W  = (const float*)d_in[1];   // (D+1,)    fp32
    const float* b  = (const float*)d_in[2];   // scalar    fp32
    float* out   = (float*)d_out;              // (T, B)    fp32
    float* xproj = (float*)d_ws;               // T*B*4 = 512 KB scratch

    dim3 block(32 * WAVES_PER_BLOCK);
    dim3 grid(ROWS / ROWS_PER_BLOCK);          // 1024 blocks, exact tiling
    xproj_wmma_kernel<<<grid, block, 0, stream>>>(x, W, xproj);
    scan_kernel<<<1, BB, 0, stream>>>(xproj, W, b, out);
}